// SpaAggregator_42975442763837
// MI455X (gfx1250) — compile-verified
//
#include <hip/hip_runtime.h>

typedef __attribute__((ext_vector_type(2))) float v2f;
typedef __attribute__((ext_vector_type(8))) float v8f;

#define KNBR   32     // neighbors per node
#define FDIM   128    // feature dim
#define EDIM   128    // embed dim
#define TILE_M 16     // output rows per block
#define LDSP   (FDIM + 4)  // padded pitch: stride 132 floats -> bank step 4, conflict-free column reads

__global__ __launch_bounds__(256) void spa_aggregator_fused(
    const float* __restrict__ id2feat,      // [N][FDIM]
    const float* __restrict__ weight,       // [FDIM][EDIM]
    const float* __restrict__ bias,         // [EDIM]
    const int*   __restrict__ neigh_idx,    // [B][KNBR]
    float*       __restrict__ out)          // [B][EDIM]
{
    __shared__ float avg[TILE_M][LDSP];     // ~8.25 KB mean-feature tile

    const int m0   = blockIdx.x * TILE_M;
    const int tid  = threadIdx.x;
    const int wave = tid >> 5;              // 0..7
    const int lane = tid & 31;

    // ---------------- Phase 1: gather + mean (HBM-bound part) ----------------
    // wave w accumulates rows 2w and 2w+1 of the tile.
    // lane l owns features [4l, 4l+4): 32 lanes x float4 = 128 floats = one 512B row.
    const float4* featBase = (const float4*)id2feat;
#pragma unroll
    for (int rr = 0; rr < 2; ++rr) {
        const int row = 2 * wave + rr;            // 0..15
        const int b   = m0 + row;
        // one neighbor index per lane, then scalar-broadcast per iteration
        const int myIdx = neigh_idx[b * KNBR + lane];
        float4 acc = make_float4(0.f, 0.f, 0.f, 0.f);
#pragma unroll
        for (int k = 0; k < KNBR; ++k) {
            const int nid = __builtin_amdgcn_readlane(myIdx, k);   // uniform -> SGPR base addr
            float4 v = featBase[(size_t)nid * (FDIM / 4) + lane];  // global_load_b128, coalesced
            acc.x += v.x; acc.y += v.y; acc.z += v.z; acc.w += v.w;
        }
        const float s = 1.0f / (float)KNBR;
        avg[row][4 * lane + 0] = acc.x * s;
        avg[row][4 * lane + 1] = acc.y * s;
        avg[row][4 * lane + 2] = acc.z * s;
        avg[row][4 * lane + 3] = acc.w * s;
    }
    __syncthreads();

    // ---------------- Phase 2: FP32 WMMA GEMM: out = avg @ W + bias ----------
    // wave w computes the 16x16 tile at columns [16w, 16w+16).
    const int n0   = wave * 16;
    const int lm   = lane & 15;    // A row / B,C column within tile
    const int half = lane >> 4;    // lane group 0/1

    v8f c = {};    // 16x16 f32 accumulator (8 VGPRs)
#pragma unroll
    for (int k = 0; k < FDIM; k += 4) {
        // A 16x4 fragment (ISA layout): lane L holds row M=L%16;
        // VGPR0 = K = k + 2*half, VGPR1 = K + 1.
        const int ka = k + 2 * half;
        v2f a;
        a.x = avg[lm][ka + 0];
        a.y = avg[lm][ka + 1];
        // B 4x16 fragment: lane L holds column N = n0 + L%16;
        // VGPR0 = row K = k + 2*half, VGPR1 = row K + 1.
        v2f bf;
        const int col = n0 + lm;
        bf.x = weight[(ka + 0) * EDIM + col];
        bf.y = weight[(ka + 1) * EDIM + col];
        // D = A x B + C   (v_wmma_f32_16x16x4_f32)
        c = __builtin_amdgcn_wmma_f32_16x16x4_f32(
                /*neg_a=*/false, a, /*neg_b=*/false, bf,
                /*c_mod=*/(short)0, c, /*reuse_a=*/false, /*reuse_b=*/false);
    }

    // Epilogue: C/D layout -> lane L, VGPR v holds (M = v + 8*half, N = lm).
    const float bv = bias[n0 + lm];
#pragma unroll
    for (int v = 0; v < 8; ++v) {
        const int row = m0 + v + 8 * half;
        out[(size_t)row * EDIM + n0 + lm] = c[v] + bv;
    }
}

extern "C" void kernel_launch(void* const* d_in, const int* in_sizes, int n_in,
                              void* d_out, int out_size, void* d_ws, size_t ws_size,
                              hipStream_t stream) {
    const float* id2feat   = (const float*)d_in[0];
    const float* weight    = (const float*)d_in[1];
    const float* bias      = (const float*)d_in[2];
    const int*   neigh_idx = (const int*)d_in[3];
    float* out = (float*)d_out;

    const int Btotal = in_sizes[3] / KNBR;      // 16384
    const int blocks = Btotal / TILE_M;         // 1024 blocks x 256 threads (8 waves)

    hipLaunchKernelGGL(spa_aggregator_fused, dim3(blocks), dim3(256), 0, stream,
                       id2feat, weight, bias, neigh_idx, out);
}